// GCNStack_87686052315400
// MI455X (gfx1250) — compile-verified
//
#include <hip/hip_runtime.h>

// GCN stack on MI455X (gfx1250, wave32).
// Bottleneck is the sparse gather/scatter (≈10 GB of traffic), not the GEMMs
// (≈40 GFLOP total), so GEMMs use the exact-precision V_WMMA_F32_16X16X4_F32.

typedef __attribute__((ext_vector_type(2))) float v2f;
typedef __attribute__((ext_vector_type(8))) float v8f;

#define D_IN 256

// ---------------- degree / normalization ----------------

__global__ void k_init_deg(float* __restrict__ deg, int n) {
  int i = blockIdx.x * blockDim.x + threadIdx.x;
  if (i < n) deg[i] = 1.0f;  // self-loop contributes 1 to every node
}

__global__ void k_count_deg(const long long* __restrict__ dst,
                            float* __restrict__ deg, int e) {
  int i = blockIdx.x * blockDim.x + threadIdx.x;
  if (i < e) unsafeAtomicAdd(&deg[(int)dst[i]], 1.0f);
}

__global__ void k_dinv(const float* __restrict__ deg,
                       float* __restrict__ dinv, int n) {
  int i = blockIdx.x * blockDim.x + threadIdx.x;
  if (i < n) {
    float d = deg[i];
    dinv[i] = (d > 0.0f) ? rsqrtf(d) : 0.0f;
  }
}

__global__ void k_zero(float* __restrict__ p, size_t nelem) {
  size_t i = (size_t)blockIdx.x * blockDim.x + threadIdx.x;
  size_t stride = (size_t)gridDim.x * blockDim.x;
  for (; i < nelem; i += stride) p[i] = 0.0f;
}

// ---------------- fp32 WMMA GEMM: C[M,N] = A[M,K] @ B[K,N] ----------------
// One wave per 16x16 output tile. M,K assumed multiples of 16/4; N guarded.
// A-frag (16x4, ISA 7.12.2):  lane holds M=lane&15, K = 2*(lane>>4)+{0,1}.
// B-frag (4x16): VGPR i: lanes 0-15 -> K=i, lanes 16-31 -> K=i+2; N=lane&15.
// C/D (16x16): VGPR i: lanes 0-15 -> M=i, lanes 16-31 -> M=i+8; N=lane&15.

__global__ void __launch_bounds__(32)
k_gemm_wmma_f32(const float* __restrict__ A, const float* __restrict__ B,
                float* __restrict__ C, int M, int N, int K) {
  const int lane  = threadIdx.x;      // 0..31, wave32
  const int row0  = blockIdx.x * 16;
  const int col0  = blockIdx.y * 16;
  const int mrow  = lane & 15;
  const int khalf = lane >> 4;        // 0 or 1
  const int bcol  = col0 + (lane & 15);
  const bool colok = bcol < N;
  const int bcol_c = colok ? bcol : (N - 1);  // clamp: keep loads in-bounds, stay branch-free

  const float* arow = A + (size_t)(row0 + mrow) * K + 2 * khalf;

  v8f acc = {};
  for (int k0 = 0; k0 < K; k0 += 4) {
    v2f a = *(const v2f*)(arow + k0);           // K-pair, contiguous -> b64 load
    int kb = k0 + 2 * khalf;
    v2f b;
    b.x = B[(size_t)kb * N + bcol_c];
    b.y = B[(size_t)(kb + 1) * N + bcol_c];
    // (neg_a, A, neg_b, B, c_mod, C, reuse_a, reuse_b)
    acc = __builtin_amdgcn_wmma_f32_16x16x4_f32(false, a, false, b,
                                                (short)0, acc, false, false);
  }

  if (colok) {
    const int rbase = row0 + khalf * 8;
    #pragma unroll
    for (int i = 0; i < 8; ++i)
      C[(size_t)(rbase + i) * N + bcol] = acc[i];
  }
}

// ---------------- sparse aggregation: agg[dst] += h[src] * norm ------------
// One block per edge, one thread per channel.

__global__ void k_aggregate(const float* __restrict__ h,
                            const long long* __restrict__ src,
                            const long long* __restrict__ dst,
                            const float* __restrict__ dinv,
                            float* __restrict__ agg, int C) {
  __shared__ int s_s, s_d;
  __shared__ float s_norm;
  const int e = blockIdx.x;
  if (threadIdx.x == 0) {
    int s = (int)src[e];
    int d = (int)dst[e];
    s_s = s;
    s_d = d;
    s_norm = dinv[s] * dinv[d];
  }
  __syncthreads();
  const int c = threadIdx.x;
  if (c < C) {
    float v = h[(size_t)s_s * C + c] * s_norm;
    unsafeAtomicAdd(&agg[(size_t)s_d * C + c], v);
  }
}

// ------------- epilogue: out = relu(agg + h_self*dinv^2 + bias) ------------

__global__ void k_finish(const float* __restrict__ agg,
                         const float* __restrict__ hself,
                         const float* __restrict__ dinv,
                         const float* __restrict__ bias,
                         float* __restrict__ out, int n, int C) {
  size_t idx = (size_t)blockIdx.x * blockDim.x + threadIdx.x;
  size_t total = (size_t)n * C;
  if (idx < total) {
    int i = (int)(idx / C);
    int c = (int)(idx % C);
    float di = dinv[i];
    float v = agg[idx] + hself[idx] * di * di + bias[c];
    out[idx] = v > 0.0f ? v : 0.0f;
  }
}

// ---------------------------------------------------------------------------

extern "C" void kernel_launch(void* const* d_in, const int* in_sizes, int n_in,
                              void* d_out, int out_size, void* d_ws, size_t ws_size,
                              hipStream_t stream) {
  const float*     x   = (const float*)d_in[0];
  const long long* ei  = (const long long*)d_in[1];  // int64 edge_index [2,E]
  const float*     W1  = (const float*)d_in[2];
  const float*     b1  = (const float*)d_in[3];
  const float*     W2  = (const float*)d_in[4];
  const float*     b2  = (const float*)d_in[5];
  float*           out = (float*)d_out;

  const int n   = in_sizes[0] / D_IN;  // 100000
  const int e   = in_sizes[1] / 2;     // 1600000
  const int dh  = in_sizes[3];         // 256
  const int dou = in_sizes[5];         // 500

  const long long* esrc = ei;
  const long long* edst = ei + e;

  // workspace partition (≈406 MB): deg | dinv | H | AGG1(->H1) | H2
  auto alignup = [](size_t v) { return (v + 255) & ~(size_t)255; };
  char* w = (char*)d_ws;
  float* deg  = (float*)w; w += alignup((size_t)n * 4);
  float* dinv = (float*)w; w += alignup((size_t)n * 4);
  float* H    = (float*)w; w += alignup((size_t)n * dh * 4);
  float* AGG1 = (float*)w; w += alignup((size_t)n * dh * 4);
  float* H2   = (float*)w; w += alignup((size_t)n * dou * 4);

  // normalization
  k_init_deg<<<(n + 255) / 256, 256, 0, stream>>>(deg, n);
  k_count_deg<<<(e + 255) / 256, 256, 0, stream>>>(edst, deg, e);
  k_dinv<<<(n + 255) / 256, 256, 0, stream>>>(deg, dinv, n);

  // ---- layer 1: H = X @ W1 ; AGG1 = scatter(edges) ; H1 = relu(...+b1) ----
  {
    dim3 grid(n / 16, dh / 16);
    k_gemm_wmma_f32<<<grid, 32, 0, stream>>>(x, W1, H, n, dh, D_IN);
  }
  size_t t1 = (size_t)n * dh;
  k_zero<<<4096, 256, 0, stream>>>(AGG1, t1);
  k_aggregate<<<e, 256, 0, stream>>>(H, esrc, edst, dinv, AGG1, dh);
  k_finish<<<(unsigned)((t1 + 255) / 256), 256, 0, stream>>>(AGG1, H, dinv, b1,
                                                             AGG1, n, dh);

  // ---- layer 2: H2 = H1 @ W2 ; out accumulates edges ; relu epilogue ------
  {
    dim3 grid(n / 16, (dou + 15) / 16);
    k_gemm_wmma_f32<<<grid, 32, 0, stream>>>(AGG1, W2, H2, n, dou, dh);
  }
  size_t t2 = (size_t)n * dou;
  k_zero<<<8192, 256, 0, stream>>>(out, t2);
  k_aggregate<<<e, 512, 0, stream>>>(H2, esrc, edst, dinv, out, dou);
  k_finish<<<(unsigned)((t2 + 255) / 256), 256, 0, stream>>>(out, H2, dinv, b2,
                                                             out, n, dou);
}